// SelfAttention2d_90598040142358
// MI455X (gfx1250) — compile-verified
//
#include <hip/hip_runtime.h>

typedef float v2f __attribute__((ext_vector_type(2)));
typedef float v8f __attribute__((ext_vector_type(8)));

#define CDIM 256
#define CK   32
#define NPIX 4096
#define NBATCH 8

// ---------------------------------------------------------------------------
// Kernel 1: per-batch Gram matrix G = X * X^T  (256x256), X = [C, N] fp32.
// One wave per 16x16 output tile, K-loop over N=4096 in steps of 4 using
// V_WMMA_F32_16X16X4_F32.  A-fragment (16x4, f32): lane l in [0,15] holds
// row M=l with K={0,1} in the VGPR pair; lanes 16..31 hold K={2,3}.
// B-fragment (4x16) mirrors it since B[k][n] = X[j0+n][k0+k].
// ---------------------------------------------------------------------------
__global__ __launch_bounds__(32) void gram_kernel(const float* __restrict__ x,
                                                  float* __restrict__ G) {
  const int b  = blockIdx.z;
  const int i0 = blockIdx.x * 16;
  const int j0 = blockIdx.y * 16;
  const float* X  = x + (size_t)b * CDIM * NPIX;
  float*       Gp = G + (size_t)b * CDIM * CDIM;

  const int lane = threadIdx.x;
  const int mn   = lane & 15;
  const int kb   = (lane >> 4) * 2;
  const int g    = lane >> 4;

  const float* arow = X + (size_t)(i0 + mn) * NPIX;
  const float* brow = X + (size_t)(j0 + mn) * NPIX;

  v8f acc = {};
  for (int k0 = 0; k0 < NPIX; k0 += 4) {
    if ((k0 & 255) == 0) {  // uniform-condition streaming prefetch
      __builtin_prefetch(arow + k0 + 256, 0, 1);
      __builtin_prefetch(brow + k0 + 256, 0, 1);
    }
    v2f a  = *reinterpret_cast<const v2f*>(arow + k0 + kb);  // 8B-aligned
    v2f bb = *reinterpret_cast<const v2f*>(brow + k0 + kb);
    acc = __builtin_amdgcn_wmma_f32_16x16x4_f32(false, a, false, bb,
                                                (short)0, acc, false, false);
  }

  // C/D layout: lanes 0-15 -> M = r, lanes 16-31 -> M = 8 + r; N = lane&15.
#pragma unroll
  for (int r = 0; r < 8; ++r) {
    const int m = r + 8 * g;
    Gp[(size_t)(i0 + m) * CDIM + (j0 + mn)] = acc[r];
  }
}

// ---------------------------------------------------------------------------
// Kernel 2: per-batch row sums s = X * 1  (one wave per row, wave32 shuffle).
// ---------------------------------------------------------------------------
__global__ __launch_bounds__(256) void rowsum_kernel(const float* __restrict__ x,
                                                     float* __restrict__ S) {
  const int b    = blockIdx.y;
  const int wave = threadIdx.x >> 5;
  const int lane = threadIdx.x & 31;
  const int row  = blockIdx.x * 8 + wave;
  const float* X = x + (size_t)b * CDIM * NPIX + (size_t)row * NPIX;

  float p = 0.f;
  for (int k = lane; k < NPIX; k += 32) p += X[k];
#pragma unroll
  for (int off = 16; off > 0; off >>= 1) p += __shfl_down(p, off, 32);
  if (lane == 0) S[b * CDIM + row] = p;
}

// ---------------------------------------------------------------------------
// Kernel 3: per-batch fold of the rank-Ck core:
//   T1 = G Wq^T                       [256 x 32]
//   M  = Wv T1 + (Wv s) bq^T + bv (Wq s)^T + N bv bq^T   [256 x 32]
//   P  = M Wk                         [256 x 256]
//   cv = M bk                         [256]
// One 256-thread block per batch; thread t owns output row t.  Weight reads
// are wave-uniform (scalar-cache broadcasts), T1/M staged through LDS.
// ---------------------------------------------------------------------------
__global__ __launch_bounds__(256) void fold_kernel(
    const float* __restrict__ Wk, const float* __restrict__ bk,
    const float* __restrict__ Wq, const float* __restrict__ bq,
    const float* __restrict__ Wv, const float* __restrict__ bv,
    const float* __restrict__ G,  const float* __restrict__ S,
    float* __restrict__ P, float* __restrict__ CV) {
  const int b = blockIdx.x;
  const float* Gp = G + (size_t)b * CDIM * CDIM;
  const float* sp = S + b * CDIM;
  float*       Pp = P + (size_t)b * CDIM * CDIM;
  float*       cp = CV + b * CDIM;
  const int t = threadIdx.x;

  __shared__ float T1[CDIM * CK];
  __shared__ float Mm[CDIM * CK];
  __shared__ float Wqs[CK];

  // Wv*s (per-thread register), Wq*s (threads 0..31 -> LDS)
  float wvs = 0.f;
  for (int c = 0; c < CDIM; ++c) wvs += Wv[t * CDIM + c] * sp[c];
  if (t < CK) {
    float a = 0.f;
    for (int c = 0; c < CDIM; ++c) a += Wq[t * CDIM + c] * sp[c];
    Wqs[t] = a;
  }

  // T1[i][k] = sum_c G[i][c] * Wq[k][c], i = t
  float acc[CK];
#pragma unroll
  for (int k = 0; k < CK; ++k) acc[k] = 0.f;
  for (int c = 0; c < CDIM; ++c) {
    const float gg = Gp[t * CDIM + c];
#pragma unroll
    for (int k = 0; k < CK; ++k) acc[k] += gg * Wq[k * CDIM + c];
  }
#pragma unroll
  for (int k = 0; k < CK; ++k) T1[t * CK + k] = acc[k];
  __syncthreads();

  // M[o][k] = sum_i Wv[o][i]*T1[i][k] + rank-1 bias terms, o = t
#pragma unroll
  for (int k = 0; k < CK; ++k) acc[k] = 0.f;
  for (int i = 0; i < CDIM; ++i) {
    const float wv = Wv[t * CDIM + i];
#pragma unroll
    for (int k = 0; k < CK; ++k) acc[k] += wv * T1[i * CK + k];
  }
  const float bvo = bv[t];
#pragma unroll
  for (int k = 0; k < CK; ++k)
    acc[k] += wvs * bq[k] + bvo * Wqs[k] + 4096.0f * bvo * bq[k];
#pragma unroll
  for (int k = 0; k < CK; ++k) Mm[t * CK + k] = acc[k];
  __syncthreads();

  // P[o][c] = sum_k M[o][k]*Wk[k][c];  cv[o] = sum_k M[o][k]*bk[k]
  float mreg[CK];
#pragma unroll
  for (int k = 0; k < CK; ++k) mreg[k] = Mm[t * CK + k];
  float cv = 0.f;
#pragma unroll
  for (int k = 0; k < CK; ++k) cv += mreg[k] * bk[k];
  cp[t] = cv;
  for (int c = 0; c < CDIM; ++c) {
    float a = 0.f;
#pragma unroll
    for (int k = 0; k < CK; ++k) a += mreg[k] * Wk[k * CDIM + c];
    Pp[t * CDIM + c] = a;
  }
}

// ---------------------------------------------------------------------------
// Kernel 4: y = gamma * (P X + cv 1^T) + x  -- the big streaming GEMM.
// One wave per 16x16 tile of [C x N]; K-loop over 256 channels (64 WMMAs).
// ---------------------------------------------------------------------------
__global__ __launch_bounds__(32) void apply_kernel(
    const float* __restrict__ x, const float* __restrict__ P,
    const float* __restrict__ CV, const float* __restrict__ gamma,
    float* __restrict__ out) {
  const int b  = blockIdx.z;
  const int h0 = blockIdx.x * 16;
  const int c0 = blockIdx.y * 16;
  const float* X  = x  + (size_t)b * CDIM * NPIX;
  const float* Pp = P  + (size_t)b * CDIM * CDIM;
  const float* cp = CV + b * CDIM;
  float*       Y  = out + (size_t)b * CDIM * NPIX;

  const int lane = threadIdx.x;
  const int mn   = lane & 15;
  const int kb   = (lane >> 4) * 2;
  const int g    = lane >> 4;

  const float* arow = Pp + (size_t)(c0 + mn) * CDIM;

  v8f acc = {};
  for (int k0 = 0; k0 < CDIM; k0 += 4) {
    v2f a = *reinterpret_cast<const v2f*>(arow + k0 + kb);
    v2f bb;
    bb.x = X[(size_t)(k0 + kb)     * NPIX + h0 + mn];  // coalesced over lanes
    bb.y = X[(size_t)(k0 + kb + 1) * NPIX + h0 + mn];
    acc = __builtin_amdgcn_wmma_f32_16x16x4_f32(false, a, false, bb,
                                                (short)0, acc, false, false);
  }

  const float gm = gamma[0];
#pragma unroll
  for (int r = 0; r < 8; ++r) {
    const int m = r + 8 * g;
    const size_t idx = (size_t)(c0 + m) * NPIX + h0 + mn;
    Y[idx] = gm * (acc[r] + cp[c0 + m]) + X[idx];
  }
}

// ---------------------------------------------------------------------------
extern "C" void kernel_launch(void* const* d_in, const int* in_sizes, int n_in,
                              void* d_out, int out_size, void* d_ws, size_t ws_size,
                              hipStream_t stream) {
  const float* x     = (const float*)d_in[0];
  const float* Wk    = (const float*)d_in[1];
  const float* bk    = (const float*)d_in[2];
  const float* Wq    = (const float*)d_in[3];
  const float* bq    = (const float*)d_in[4];
  const float* Wv    = (const float*)d_in[5];
  const float* bv    = (const float*)d_in[6];
  const float* gamma = (const float*)d_in[7];
  float* out = (float*)d_out;

  // Workspace layout (fp32): G[8][256][256], s[8][256], P[8][256][256], cv[8][256]
  float* wsf = (float*)d_ws;
  float* G  = wsf;
  float* S  = G + (size_t)NBATCH * CDIM * CDIM;
  float* P  = S + NBATCH * CDIM;
  float* CV = P + (size_t)NBATCH * CDIM * CDIM;

  gram_kernel  <<<dim3(16, 16, NBATCH), 32,  0, stream>>>(x, G);
  rowsum_kernel<<<dim3(32, NBATCH),     256, 0, stream>>>(x, S);
  fold_kernel  <<<dim3(NBATCH),         256, 0, stream>>>(Wk, bk, Wq, bq, Wv, bv,
                                                          G, S, P, CV);
  apply_kernel <<<dim3(NPIX / 16, CDIM / 16, NBATCH), 32, 0, stream>>>(x, P, CV,
                                                                       gamma, out);
}